// GNN_6786048328010
// MI455X (gfx1250) — compile-verified
//
#include <hip/hip_runtime.h>

#define N_NODES 100000
#define N_EDGES 1600000
#define HIDDEN  64
#define KDIM    128          // fused K: [agg | x]
#define A_STRIDE 132         // 128 + 4 pad -> conflict-free A-fragment reads

typedef __attribute__((ext_vector_type(2))) float v2f;
typedef __attribute__((ext_vector_type(8))) float v8f;

// ---------------- utility: zero a float buffer ----------------
__global__ void zero_f32(float* __restrict__ p, int n) {
    int i = blockIdx.x * blockDim.x + threadIdx.x;
    int stride = gridDim.x * blockDim.x;
    for (; i < n; i += stride) p[i] = 0.0f;
}

// ---------------- degree: deg[d] += 1 per edge ----------------
__global__ void degree_kernel(const int* __restrict__ dst, float* __restrict__ deg) {
    int e = blockIdx.x * blockDim.x + threadIdx.x;
    if (e < N_EDGES) atomicAdd(&deg[dst[e]], 1.0f);
}

// ------- scatter: agg[dst] += h[src], 16 threads/edge, float4 each -------
__global__ void scatter_kernel(const float* __restrict__ h,
                               const int* __restrict__ src,
                               const int* __restrict__ dst,
                               float* __restrict__ agg) {
    int idx = blockIdx.x * blockDim.x + threadIdx.x;
    int e = idx >> 4;
    int q = (idx & 15) << 2;          // column group of 4
    if (e >= N_EDGES) return;
    int s = src[e];
    int d = dst[e];
    const float4 v = *(const float4*)(h + (size_t)s * HIDDEN + q);
    float* a = agg + (size_t)d * HIDDEN + q;
    atomicAdd(a + 0, v.x);
    atomicAdd(a + 1, v.y);
    atomicAdd(a + 2, v.z);
    atomicAdd(a + 3, v.w);
}

// ------- fused: out = ELU( (agg/deg) @ Wl + b + x @ Wr ) via f32 WMMA -------
// block: 256 thr = 8 waves; block owns 32 nodes; wave -> one 16x16 C tile.
__global__ __launch_bounds__(256)
void sage_gemm_kernel(const float* __restrict__ agg,
                      const float* __restrict__ deg,
                      const float* __restrict__ xin,
                      const float* __restrict__ Wl,
                      const float* __restrict__ bl,
                      const float* __restrict__ Wr,
                      float* __restrict__ out) {
    __shared__ float ldsA[32 * A_STRIDE];       // [32 nodes][128 K] padded
    __shared__ float ldsW[KDIM * HIDDEN];       // [128][64]: rows 0-63 Wl, 64-127 Wr

    const int tid = threadIdx.x;
    const int nodeBase = blockIdx.x * 32;

    // stage combined weights into LDS (2048 float4 / 256 threads = 8 each)
    #pragma unroll
    for (int i = 0; i < 8; ++i) {
        int flat = (tid + i * 256) * 4;
        float4 w;
        if (flat < HIDDEN * HIDDEN) w = *(const float4*)(Wl + flat);
        else                        w = *(const float4*)(Wr + (flat - HIDDEN * HIDDEN));
        *(float4*)(ldsW + flat) = w;
    }

    // stage A tile: cols 0-63 = agg/max(deg,1), cols 64-127 = x   (1024 float4)
    #pragma unroll
    for (int i = 0; i < 4; ++i) {
        int j = tid + i * 256;
        int row = j >> 5;
        int col = (j & 31) << 2;
        int node = nodeBase + row;                 // N_NODES % 32 == 0: always valid
        float4 v;
        if (col < HIDDEN) {
            v = *(const float4*)(agg + (size_t)node * HIDDEN + col);
            float rd = 1.0f / fmaxf(deg[node], 1.0f);
            v.x *= rd; v.y *= rd; v.z *= rd; v.w *= rd;
        } else {
            v = *(const float4*)(xin + (size_t)node * HIDDEN + (col - HIDDEN));
        }
        *(float4*)(ldsA + row * A_STRIDE + col) = v;
    }

    __syncthreads();

    const int lane    = tid & 31;
    const int wave    = tid >> 5;
    const int rowTile = wave >> 2;          // 0..1
    const int colTile = wave & 3;           // 0..3
    const int m       = lane & 15;          // A row / B col within tile
    const int khi     = (lane >> 4) << 1;   // 0 or 2 (K sub-offset per ISA layout)

    const float* aRow = ldsA + (rowTile * 16 + m) * A_STRIDE + khi;
    const float* bCol = ldsW + khi * HIDDEN + colTile * 16 + m;

    v8f c = {0.f, 0.f, 0.f, 0.f, 0.f, 0.f, 0.f, 0.f};
    #pragma unroll
    for (int k = 0; k < KDIM; k += 4) {
        v2f a, b;
        a.x = aRow[k];
        a.y = aRow[k + 1];
        b.x = bCol[k * HIDDEN];
        b.y = bCol[(k + 1) * HIDDEN];
        // D = A(16x4,f32) * B(4x16,f32) + C(16x16,f32)
        c = __builtin_amdgcn_wmma_f32_16x16x4_f32(
                /*neg_a=*/false, a, /*neg_b=*/false, b,
                /*c_mod=*/(short)0, c, /*reuse_a=*/false, /*reuse_b=*/false);
    }

    // epilogue: bias + ELU, store. C VGPR i: M = i + 8*(lane>=16), N = lane&15
    const int colBase = colTile * 16;
    const float bias  = bl[colBase + m];
    const int mOff    = (lane >> 4) << 3;   // 0 or 8
    float* outBase = out + (size_t)(nodeBase + rowTile * 16 + mOff) * HIDDEN + colBase + m;
    #pragma unroll
    for (int i = 0; i < 8; ++i) {
        float v = c[i] + bias;
        v = (v > 0.0f) ? v : (__expf(v) - 1.0f);   // ELU, alpha = 1
        outBase[i * HIDDEN] = v;
    }
}

// ---------------------------------------------------------------------------
extern "C" void kernel_launch(void* const* d_in, const int* in_sizes, int n_in,
                              void* d_out, int out_size, void* d_ws, size_t ws_size,
                              hipStream_t stream) {
    const float* x   = (const float*)d_in[0];
    const int*   ei  = (const int*)  d_in[1];
    const float* Wl1 = (const float*)d_in[2];
    const float* bl1 = (const float*)d_in[3];
    const float* Wr1 = (const float*)d_in[4];
    const float* Wl2 = (const float*)d_in[5];
    const float* bl2 = (const float*)d_in[6];
    const float* Wr2 = (const float*)d_in[7];
    const float* Wl3 = (const float*)d_in[8];
    const float* bl3 = (const float*)d_in[9];
    const float* Wr3 = (const float*)d_in[10];

    const int* src = ei;
    const int* dst = ei + N_EDGES;

    float* ws  = (float*)d_ws;
    float* deg = ws;                                   // 100000 (pad to 102400)
    float* agg = ws + 102400;                          // 6.4M
    float* h1  = agg + (size_t)N_NODES * HIDDEN;       // 6.4M
    float* h2  = h1  + (size_t)N_NODES * HIDDEN;       // 6.4M
    float* outp = (float*)d_out;

    const dim3 blk(256);

    // degree (recomputed every call: deterministic, no cross-call state)
    zero_f32<<<1024, blk, 0, stream>>>(deg, N_NODES);
    degree_kernel<<<(N_EDGES + 255) / 256, blk, 0, stream>>>(dst, deg);

    const float* Wls[3] = {Wl1, Wl2, Wl3};
    const float* bls[3] = {bl1, bl2, bl3};
    const float* Wrs[3] = {Wr1, Wr2, Wr3};
    float*       hout[3] = {h1, h2, outp};

    const float* hin = x;
    for (int l = 0; l < 3; ++l) {
        zero_f32<<<4096, blk, 0, stream>>>(agg, N_NODES * HIDDEN);
        scatter_kernel<<<(N_EDGES * 16) / 256, blk, 0, stream>>>(hin, src, dst, agg);
        sage_gemm_kernel<<<N_NODES / 32, blk, 0, stream>>>(
            agg, deg, hin, Wls[l], bls[l], Wrs[l], hout[l]);
        hin = hout[l];
    }
}